// AttnGraphConvolution_12232066859196
// MI455X (gfx1250) — compile-verified
//
#include <hip/hip_runtime.h>
#include <math.h>

// ---------------------------------------------------------------------------
// AttnGraphConvolution (GAT layer) for gfx1250.
// adj stream (268 MB) -> HBM bound, ~11.5us floor at 23.3 TB/s.
//   K1: h = x @ W                    (fp32 WMMA 16x16x4)
//   K2: hswz = WMMA-B-layout pack of h (so B operands load as b128)
//   K3: s1 = h@a1, s2 = h@a2
//   K4: fused masked-softmax attention, column-split x4, partials to ws
//   K5: reduce partials, normalize, ELU
// ---------------------------------------------------------------------------

#define N_NODES 8192
#define IN_F    256
#define OUT_F   128
#define ALPHA   0.2f
#define NBLK    (N_NODES / 16)     // 512 row blocks
#define NSPLIT  4                  // column splits
#define CSPLIT  (N_NODES / NSPLIT) // 2048 columns per split

typedef __attribute__((ext_vector_type(2))) float v2f;
typedef __attribute__((ext_vector_type(4))) float v4f;
typedef __attribute__((ext_vector_type(8))) float v8f;

// -------------------------------------------------------------------
// Kernel 1: h = x @ W.  One wave per 16x16 tile of h.
// A (16x4 f32): lane&15 = row M; lane half selects K pair {0,1}/{2,3}.
// B (4x16 f32): lane half selects K pair, lane&15 = column N.
// -------------------------------------------------------------------
__global__ void k_h_gemm(const float* __restrict__ x,
                         const float* __restrict__ W,
                         float* __restrict__ h) {
  const int tile = blockIdx.x;                 // 512 * 8 tiles
  const int tn   = tile & 7;
  const int tm   = tile >> 3;
  const int lane = threadIdx.x & 31;
  const int col  = lane & 15;
  const int koff = (lane >> 4) << 1;

  v8f acc = {};
  const float* xrow = x + (size_t)(tm * 16 + col) * IN_F;
  const float* wcol = W + tn * 16 + col;

  for (int k = 0; k < IN_F; k += 4) {
    v2f A;
    A.x = xrow[k + koff];
    A.y = xrow[k + koff + 1];
    const float* wp = wcol + (size_t)(k + koff) * OUT_F;
    v2f B;
    B.x = wp[0];
    B.y = wp[OUT_F];
    acc = __builtin_amdgcn_wmma_f32_16x16x4_f32(
        false, A, false, B, (short)0, acc, false, false);
  }

  const int rbase = tm * 16 + ((lane >> 4) << 3);
  float* hp = h + tn * 16 + col;
#pragma unroll
  for (int j = 0; j < 8; ++j) {
    hp[(size_t)(rbase + j) * OUT_F] = acc[j];
  }
}

// -------------------------------------------------------------------
// Kernel 2: pack h into WMMA-B-operand order so that in the attention
// kernel each lane's 16 B floats per 4-column K-step are contiguous.
// dest index d decodes as: v=d&1, t=(d>>1)&7, ncol=(d>>4)&15,
// khalf=(d>>8)&1, blk=d>>9 ; source row r = blk*4+khalf*2+v, n=t*16+ncol.
// -------------------------------------------------------------------
__global__ void k_swz(const float* __restrict__ h, float* __restrict__ hswz) {
  const int d = blockIdx.x * blockDim.x + threadIdx.x;   // 1M elements
  const int v    = d & 1;
  const int t    = (d >> 1) & 7;
  const int ncol = (d >> 4) & 15;
  const int kh   = (d >> 8) & 1;
  const int blk  = d >> 9;
  const int r = blk * 4 + kh * 2 + v;
  const int n = t * 16 + ncol;
  hswz[d] = h[(size_t)r * OUT_F + n];
}

// -------------------------------------------------------------------
// Kernel 3: s1[i] = h[i,:].a1 ; s2[i] = h[i,:].a2  (one wave per row)
// -------------------------------------------------------------------
__global__ void k_scores(const float* __restrict__ h,
                         const float* __restrict__ a,
                         float* __restrict__ s1,
                         float* __restrict__ s2) {
  const int lane = threadIdx.x & 31;
  const int row  = blockIdx.x * (blockDim.x >> 5) + (threadIdx.x >> 5);
  if (row >= N_NODES) return;

  const float* hr = h + (size_t)row * OUT_F;
  float p1 = 0.f, p2 = 0.f;
#pragma unroll
  for (int c = 0; c < OUT_F; c += 32) {
    float v = hr[c + lane];
    p1 += v * a[c + lane];
    p2 += v * a[OUT_F + c + lane];
  }
#pragma unroll
  for (int off = 16; off > 0; off >>= 1) {
    p1 += __shfl_down(p1, off);
    p2 += __shfl_down(p2, off);
  }
  if (lane == 0) {
    s1[row] = p1;
    s2[row] = p2;
  }
}

// -------------------------------------------------------------------
// Kernel 4: fused attention partial pass. grid = (512 row blocks,
// 4 column splits); one wave each, 16 rows x 2048 columns.
// p = adj>0 ? exp(lrelu(s1+s2)) : 0  (no max shift: |e| <~ 12, exp is
// safe in fp32; diagonal always present so denominator > 0).
// acc(16x128) += P(16x4) @ H(4x128), 8 fp32 WMMAs per K-step.
// adj: non-temporal b64 loads + prefetch. hswz B operands: 4x b128.
// -------------------------------------------------------------------
__global__ void k_attn(const float* __restrict__ adj,
                       const float* __restrict__ hswz,
                       const float* __restrict__ s1,
                       const float* __restrict__ s2,
                       float* __restrict__ pacc,
                       float* __restrict__ plsum) {
  const int rb    = blockIdx.x;          // row block
  const int split = blockIdx.y;          // column split
  const int i0    = rb * 16;
  const int cbeg  = split * CSPLIT;
  const int cend  = cbeg + CSPLIT;
  const int lane  = threadIdx.x & 31;
  const int col   = lane & 15;           // A row M ; B/C/D column N
  const int khalf = lane >> 4;
  const int koff  = khalf << 1;

  v8f acc[8];
#pragma unroll
  for (int t = 0; t < 8; ++t) acc[t] = (v8f){0, 0, 0, 0, 0, 0, 0, 0};
  float lsum = 0.f;

  const float  s1m    = s1[i0 + col];
  const float* adjrow = adj + (size_t)(i0 + col) * N_NODES;

  for (int c = cbeg; c < cend; c += 4) {
    if ((c & 63) == 0) {
      __builtin_prefetch(adjrow + c + 2048, 0, 0);   // 8KB ahead in this row
    }
    const v2f av =
        __builtin_nontemporal_load(reinterpret_cast<const v2f*>(adjrow + c) + khalf);
    const v2f sv = *reinterpret_cast<const v2f*>(s2 + c + koff);

    float e0 = s1m + sv.x;
    float e1 = s1m + sv.y;
    e0 = e0 > 0.f ? e0 : ALPHA * e0;
    e1 = e1 > 0.f ? e1 : ALPHA * e1;
    const float p0 = av.x > 0.f ? __expf(e0) : 0.f;
    const float p1 = av.y > 0.f ? __expf(e1) : 0.f;
    lsum += p0 + p1;

    v2f A;
    A.x = p0;
    A.y = p1;

    // B operands: 16 contiguous floats for this lane -> 4x b128
    const v4f* hq = reinterpret_cast<const v4f*>(
        hswz + ((size_t)(c >> 2) * 2 + khalf) * 256 + col * 16);
    v4f q[4];
    q[0] = hq[0];
    q[1] = hq[1];
    q[2] = hq[2];
    q[3] = hq[3];
#pragma unroll
    for (int t = 0; t < 8; ++t) {
      v2f B;
      B.x = q[t >> 1][(t & 1) * 2];
      B.y = q[t >> 1][(t & 1) * 2 + 1];
      acc[t] = __builtin_amdgcn_wmma_f32_16x16x4_f32(
          false, A, false, B, (short)0, acc[t], false, false);
    }
  }

  // lanes L and L+16 both accumulated row L&15 -> combine halves
  lsum += __shfl_xor(lsum, 16);

  // write partial 16x128 tile + partial row sums
  float* pa = pacc + (size_t)(split * NBLK + rb) * (16 * OUT_F);
#pragma unroll
  for (int j = 0; j < 8; ++j) {
    const int row = j + (khalf << 3);
#pragma unroll
    for (int t = 0; t < 8; ++t) {
      pa[row * OUT_F + 16 * t + col] = acc[t][j];
    }
  }
  if (lane < 16) {
    plsum[(size_t)(split * NBLK + rb) * 16 + col] = lsum;
  }
}

// -------------------------------------------------------------------
// Kernel 5: reduce the 4 column-split partials, normalize, ELU, store.
// One thread per output element (tid == r*128 + n).
// -------------------------------------------------------------------
__global__ void k_reduce(const float* __restrict__ pacc,
                         const float* __restrict__ plsum,
                         float* __restrict__ out) {
  const int tid = blockIdx.x * blockDim.x + threadIdx.x;  // 1M
  const int n  = tid & (OUT_F - 1);
  const int r  = tid >> 7;
  const int rb = r >> 4;
  const int rl = r & 15;

  float sum = 0.f, l = 0.f;
#pragma unroll
  for (int s = 0; s < NSPLIT; ++s) {
    sum += pacc[((size_t)(s * NBLK + rb) * 16 + rl) * OUT_F + n];
    l   += plsum[(size_t)(s * NBLK + rb) * 16 + rl];
  }
  float v = sum / l;
  v = v > 0.f ? v : (__expf(v) - 1.0f);   // ELU (alpha = 1)
  out[tid] = v;
}

// -------------------------------------------------------------------
// Launcher. Inputs: x (8192x256), adj (8192x8192), W (256x128), a (256).
// Output: 8192x128 fp32.
// Workspace: h 4MB | hswz 4MB | s1 32KB | s2 32KB | plsum 128KB | pacc 16MB
// -------------------------------------------------------------------
extern "C" void kernel_launch(void* const* d_in, const int* in_sizes, int n_in,
                              void* d_out, int out_size, void* d_ws, size_t ws_size,
                              hipStream_t stream) {
  const float* x   = (const float*)d_in[0];
  const float* adj = (const float*)d_in[1];
  const float* W   = (const float*)d_in[2];
  const float* a   = (const float*)d_in[3];
  float* out = (float*)d_out;

  float* h     = (float*)d_ws;                          // 8192*128
  float* hswz  = h + (size_t)N_NODES * OUT_F;           // 8192*128
  float* s1    = hswz + (size_t)N_NODES * OUT_F;        // 8192
  float* s2    = s1 + N_NODES;                          // 8192
  float* plsum = s2 + N_NODES;                          // 4*512*16
  float* pacc  = plsum + (size_t)NSPLIT * NBLK * 16;    // 4*512*16*128

  // K1: h = x @ W
  k_h_gemm<<<dim3((N_NODES / 16) * (OUT_F / 16)), dim3(32), 0, stream>>>(x, W, h);

  // K2: pack h into WMMA B-operand order
  k_swz<<<dim3((N_NODES * OUT_F) / 256), dim3(256), 0, stream>>>(h, hswz);

  // K3: attention scores
  k_scores<<<dim3(N_NODES / 8), dim3(256), 0, stream>>>(h, a, s1, s2);

  // K4: fused masked softmax attention partials (512 row blocks x 4 splits)
  k_attn<<<dim3(NBLK, NSPLIT), dim3(32), 0, stream>>>(adj, hswz, s1, s2, pacc, plsum);

  // K5: combine partials, normalize, ELU
  k_reduce<<<dim3((N_NODES * OUT_F) / 256), dim3(256), 0, stream>>>(pacc, plsum, out);
}